// CARAFE_8169027797569
// MI455X (gfx1250) — compile-verified
//
#include <hip/hip_runtime.h>
#include <math.h>

// ---------------------------------------------------------------------------
// CARAFE on gfx1250: two WMMA (f32 16x16x4) GEMM kernels for the conv stages,
// plus a bandwidth-bound softmax+reassembly kernel on the low-res grid.
// ---------------------------------------------------------------------------

typedef __attribute__((ext_vector_type(2))) float v2f;
typedef __attribute__((ext_vector_type(8))) float v8f;

#define EPSV 1e-5f

// ---------------------------------------------------------------------------
// Kernel 1: comp = ReLU(BN(1x1 conv, 256 -> 64))
// GEMM per batch: D[64 x 4096] = Wc[64 x 256] * X[256 x 4096]
// One wave computes a 64(cm) x 16(pixel) output tile: 4 WMMA acc tiles.
// blockDim = 64 (2 waves). 256 blocks -> 512 waves = 2 batches * 256 tiles.
// ---------------------------------------------------------------------------
__global__ __launch_bounds__(64) void carafe_conv1_wmma(
    const float* __restrict__ x,        // [2,256,64,64]
    const float* __restrict__ comp_w,   // [64,256]
    const float* __restrict__ gamma, const float* __restrict__ beta,
    const float* __restrict__ mean,  const float* __restrict__ var,
    float* __restrict__ w1)             // [2,64,4096]
{
    __shared__ float2 sb[64];           // per-channel BN scale/bias
    int t = threadIdx.x;
    if (t < 64) {
        float sc = gamma[t] * rsqrtf(var[t] + EPSV);
        sb[t] = make_float2(sc, beta[t] - mean[t] * sc);
    }
    __syncthreads();

    int lane = t & 31;
    int lm   = lane & 15;        // n / m index within half
    int half = lane >> 4;        // 0 or 1
    int gw   = blockIdx.x * 2 + (t >> 5);     // global wave id 0..511
    int b    = gw >> 8;                       // batch
    int p0   = (gw & 255) << 4;               // pixel tile start

    const float* xb = x + (size_t)b * 256 * 4096;

    v8f acc[4] = {};
    for (int k0 = 0; k0 < 256; k0 += 4) {
        int kA = k0 + half * 2;              // k for vgpr v: kA + v
        // B tile: B[k][n] = x[b, k, p0+lm]
        v2f bv;
        bv.x = xb[(size_t)(kA + 0) * 4096 + p0 + lm];
        bv.y = xb[(size_t)(kA + 1) * 4096 + p0 + lm];
#pragma unroll
        for (int tt = 0; tt < 4; tt++) {
            // A tile: A[m][k] = comp_w[(tt*16+lm)*256 + kA + v]
            const float* ap = comp_w + (size_t)(tt * 16 + lm) * 256 + kA;
            v2f av; av.x = ap[0]; av.y = ap[1];
            acc[tt] = __builtin_amdgcn_wmma_f32_16x16x4_f32(
                false, av, false, bv, (short)0, acc[tt], false, false);
        }
    }

    // BN + ReLU + store.  D layout: lane,vgpr v -> m = tt*16 + v + 8*half, n = lm
    float* w1b = w1 + (size_t)b * 64 * 4096;
#pragma unroll
    for (int tt = 0; tt < 4; tt++) {
#pragma unroll
        for (int v = 0; v < 8; v++) {
            int cm = tt * 16 + v + 8 * half;
            float2 s = sb[cm];
            float val = fmaf(acc[tt][v], s.x, s.y);
            w1b[(size_t)cm * 4096 + p0 + lm] = fmaxf(val, 0.0f);
        }
    }
}

// ---------------------------------------------------------------------------
// Kernel 2: enc = BN(3x3 conv, 64 -> 100), pad 1. im2col on the fly.
// GEMM per batch: D[100 x 4096] = We[100 x 576] * Im2col[576 x 4096]
// One wave: 112(cm, padded) x 16(pixel) tile -> 7 WMMA acc tiles.
// ---------------------------------------------------------------------------
__device__ __forceinline__ float im2col_val(const float* __restrict__ w1b,
                                            int k, int py, int px)
{
    int c  = k / 9;
    int r  = k - c * 9;
    int dy = r / 3;
    int dx = r - dy * 3;
    int y  = py + dy - 1;
    int xx = px + dx - 1;
    if ((unsigned)y < 64u && (unsigned)xx < 64u)
        return w1b[(size_t)c * 4096 + y * 64 + xx];
    return 0.0f;
}

__global__ __launch_bounds__(64) void carafe_conv2_wmma(
    const float* __restrict__ w1,       // [2,64,4096]
    const float* __restrict__ enc_w,    // [100,576]
    const float* __restrict__ gamma, const float* __restrict__ beta,
    const float* __restrict__ mean,  const float* __restrict__ var,
    float* __restrict__ w2)             // [2,100,4096]
{
    __shared__ float2 sb[100];
    int t = threadIdx.x;
    for (int c = t; c < 100; c += 64) {
        float sc = gamma[c] * rsqrtf(var[c] + EPSV);
        sb[c] = make_float2(sc, beta[c] - mean[c] * sc);
    }
    __syncthreads();

    int lane = t & 31;
    int lm   = lane & 15;
    int half = lane >> 4;
    int gw   = blockIdx.x * 2 + (t >> 5);     // 0..511
    int b    = gw >> 8;
    int p0   = (gw & 255) << 4;
    int py   = p0 >> 6;
    int px   = (p0 & 63) + lm;                // row fixed per wave (64 % 16 == 0)

    const float* w1b = w1 + (size_t)b * 64 * 4096;

    v8f acc[7] = {};
    for (int k0 = 0; k0 < 576; k0 += 4) {
        int kA = k0 + half * 2;
        v2f bv;
        bv.x = im2col_val(w1b, kA + 0, py, px);
        bv.y = im2col_val(w1b, kA + 1, py, px);
#pragma unroll
        for (int tt = 0; tt < 7; tt++) {
            int m = tt * 16 + lm;
            const float* ap = enc_w + (size_t)(m < 100 ? m : 0) * 576 + kA;
            float a0 = ap[0], a1 = ap[1];
            if (m >= 100) { a0 = 0.0f; a1 = 0.0f; }
            v2f av; av.x = a0; av.y = a1;
            acc[tt] = __builtin_amdgcn_wmma_f32_16x16x4_f32(
                false, av, false, bv, (short)0, acc[tt], false, false);
        }
    }

    float* w2b = w2 + (size_t)b * 100 * 4096;
#pragma unroll
    for (int tt = 0; tt < 7; tt++) {
#pragma unroll
        for (int v = 0; v < 8; v++) {
            int cm = tt * 16 + v + 8 * half;
            if (cm < 100) {
                float2 s = sb[cm];
                w2b[(size_t)cm * 4096 + p0 + lm] = fmaf(acc[tt][v], s.x, s.y);
            }
        }
    }
}

// ---------------------------------------------------------------------------
// Kernel 3: pixel-shuffle + softmax(25) + content-aware reassembly.
// Collapsed to the low-res grid:
//   out[b,c,2h+i,2w+j] = sum_{ki,kj} W[b,(5ki+kj)*4+(2i+j),h,w]
//                                     * x[b,c,h+ki-2,w+kj-2]
// One block per (b,h) row; lanes over w (coalesced), loop channels.
// ---------------------------------------------------------------------------
__global__ __launch_bounds__(256) void carafe_reassemble(
    const float* __restrict__ x,    // [2,256,64,64]
    const float* __restrict__ w2,   // [2,100,4096]
    float* __restrict__ out)        // [2,256,128,128]
{
    __shared__ float wsm[64][105];  // [w][s*26 + k]; odd row stride: no bank conflicts
    int t = threadIdx.x;
    int b = blockIdx.x >> 6;
    int h = blockIdx.x & 63;

    // Phase 1: 64 w-positions x 4 sub-positions -> softmax over 25 taps
    {
        int w = t & 63;
        int s = t >> 6;                          // s = i*2 + j
        const float* p = w2 + (size_t)b * 100 * 4096 + h * 64 + w;
        float v[25];
        float mx = -3.0e38f;
#pragma unroll
        for (int k = 0; k < 25; k++) {
            v[k] = p[(size_t)(k * 4 + s) * 4096];
            mx = fmaxf(mx, v[k]);
        }
        float sum = 0.0f;
#pragma unroll
        for (int k = 0; k < 25; k++) { v[k] = expf(v[k] - mx); sum += v[k]; }
        float inv = 1.0f / sum;
#pragma unroll
        for (int k = 0; k < 25; k++) wsm[w][s * 26 + k] = v[k] * inv;
    }
    __syncthreads();

    // Phase 2: lanes over w, 4 channels at a time
    int w  = t & 63;
    int cq = t >> 6;
    const float* xb = x + (size_t)b * 256 * 4096;
    float* ob = out + (size_t)b * 256 * 16384;

    for (int c0 = 0; c0 < 256; c0 += 4) {
        int c = c0 + cq;
        const float* xc = xb + (size_t)c * 4096;
        float xv[25];
#pragma unroll
        for (int ki = 0; ki < 5; ki++) {
            int y = h + ki - 2;
            bool yok = (unsigned)y < 64u;
#pragma unroll
            for (int kj = 0; kj < 5; kj++) {
                int xx = w + kj - 2;
                xv[ki * 5 + kj] =
                    (yok && (unsigned)xx < 64u) ? xc[y * 64 + xx] : 0.0f;
            }
        }
#pragma unroll
        for (int s = 0; s < 4; s++) {
            float o = 0.0f;
#pragma unroll
            for (int k = 0; k < 25; k++) o = fmaf(wsm[w][s * 26 + k], xv[k], o);
            int i = s >> 1, j = s & 1;
            ob[(size_t)c * 16384 + (2 * h + i) * 128 + (2 * w + j)] = o;
        }
    }
}

// ---------------------------------------------------------------------------
extern "C" void kernel_launch(void* const* d_in, const int* in_sizes, int n_in,
                              void* d_out, int out_size, void* d_ws, size_t ws_size,
                              hipStream_t stream) {
    const float* x          = (const float*)d_in[0];
    const float* comp_w     = (const float*)d_in[1];
    const float* comp_gamma = (const float*)d_in[2];
    const float* comp_beta  = (const float*)d_in[3];
    const float* comp_mean  = (const float*)d_in[4];
    const float* comp_var   = (const float*)d_in[5];
    const float* enc_w      = (const float*)d_in[6];
    const float* enc_gamma  = (const float*)d_in[7];
    const float* enc_beta   = (const float*)d_in[8];
    const float* enc_mean   = (const float*)d_in[9];
    const float* enc_var    = (const float*)d_in[10];

    float* W1 = (float*)d_ws;                 // 2*64*4096  floats (2 MB)
    float* W2 = W1 + 2 * 64 * 4096;           // 2*100*4096 floats (3.2 MB)

    carafe_conv1_wmma<<<256, 64, 0, stream>>>(
        x, comp_w, comp_gamma, comp_beta, comp_mean, comp_var, W1);
    carafe_conv2_wmma<<<256, 64, 0, stream>>>(
        W1, enc_w, enc_gamma, enc_beta, enc_mean, enc_var, W2);
    carafe_reassemble<<<128, 256, 0, stream>>>(x, W2, (float*)d_out);
}